// QuanvolutionNet_65481071408502
// MI455X (gfx1250) — compile-verified
//
#include <hip/hip_runtime.h>
#include <hip/hip_bf16.h>
#include <math.h>

// MI455X / gfx1250, wave32. Memory-bound workload (~26 MB dominant read of x).
// Quantum sim fully in registers; classifier head on V_WMMA_F32_16X16X4_F32.

typedef float v2f __attribute__((ext_vector_type(2)));
typedef float v8f __attribute__((ext_vector_type(8)));

#define B_TOTAL 8192
#define POS 196          // 14x14 patch positions
#define BPB 16           // batch rows per block in main kernel (== WMMA M)
#define NBLK (B_TOTAL / BPB)

// ---------------- kernel 0: init per-position min/max (ws is poisoned) -------
__global__ void qv_init(int* __restrict__ mnb, int* __restrict__ mxb) {
  int t = blockIdx.x * blockDim.x + threadIdx.x;
  if (t < POS) {
    mnb[t] = 0x7F800000;  // +inf bits
    mxb[t] = 0;           // 0.0f bits (all inputs are >= 0)
  }
}

// ---------------- kernel 1: global per-position min/max over batch -----------
// x in [0,1) => non-negative => IEEE float bit pattern order == signed-int
// order, so integer atomicMin/Max give a deterministic float min/max.
__global__ __launch_bounds__(256) void qv_minmax(const float* __restrict__ x,
                                                 int* __restrict__ mnb,
                                                 int* __restrict__ mxb) {
  __shared__ int smn[POS], smx[POS];
  const int tid = threadIdx.x;
  for (int i = tid; i < POS; i += 256) { smn[i] = 0x7F800000; smx[i] = 0; }
  __syncthreads();
  const int b0 = blockIdx.x * 128;  // 64 blocks x 128 batches
  for (int idx = tid; idx < 128 * 784; idx += 256) {  // fully coalesced
    int bl = idx / 784, pix = idx - bl * 784;
    int r = pix / 28, c = pix - r * 28;
    int p = (r >> 1) * 14 + (c >> 1);
    int v = __float_as_int(x[(b0 + bl) * 784 + pix]);
    atomicMin(&smn[p], v);
    atomicMax(&smx[p], v);
  }
  __syncthreads();
  for (int i = tid; i < POS; i += 256) {
    atomicMin(&mnb[i], smn[i]);
    atomicMax(&mxb[i], smx[i]);
  }
}

// ---------------- kernel 2: fused quanvolution + residual + WMMA head --------
__global__ __launch_bounds__(256) void qv_main(
    const float* __restrict__ x, const float* __restrict__ params,
    const float* __restrict__ res_w, const float* __restrict__ cls_w,
    const float* __restrict__ cls_b, const float* __restrict__ mn,
    const float* __restrict__ mx, float* __restrict__ out) {
  __shared__ float feat[BPB][788];  // 784 features, stride 788 -> 16 distinct banks for column reads
  __shared__ float lgt[BPB][16];
  const int tid = threadIdx.x;
  const int b0 = blockIdx.x * BPB;

  // shared variational angles (3 layers x 4 qubits), half-angle cos/sin
  float lc[12], ls[12];
#pragma unroll
  for (int i = 0; i < 12; ++i) {
    float th = 0.5f * params[i];
    lc[i] = __cosf(th);
    ls[i] = __sinf(th);
  }

  // ---- 16 x 196 quantum sims, 16-amplitude state vector in registers ----
  for (int s = tid; s < BPB * POS; s += 256) {
    int bi = s / POS;
    int p = s - bi * POS;
    int prow = p / 14, pcol = p - prow * 14;
    const float* ip = x + (b0 + bi) * 784 + (2 * prow) * 28 + 2 * pcol;
    float e0 = ip[0], e1 = ip[1], e2 = ip[28], e3 = ip[29];
    float lo = mn[p];
    // 0.5 * pi / (max - min + 1e-8): angle scaling folded with half-angle
    float sc = 1.57079632679489662f / (mx[p] - lo + 1e-8f);
    float a0 = (e0 - lo) * sc, a1 = (e1 - lo) * sc;
    float a2 = (e2 - lo) * sc, a3 = (e3 - lo) * sc;
    float c0 = __cosf(a0), s0 = __sinf(a0);
    float c1 = __cosf(a1), s1 = __sinf(a1);
    float c2 = __cosf(a2), s2 = __sinf(a2);
    float c3 = __cosf(a3), s3 = __sinf(a3);

    // product state after input RYs on |0000>; qubit q has bit weight 8>>q
    float st[16];
#pragma unroll
    for (int i = 0; i < 16; ++i) {
      st[i] = ((i & 8) ? s0 : c0) * ((i & 4) ? s1 : c1) *
              ((i & 2) ? s2 : c2) * ((i & 1) ? s3 : c3);
    }
#pragma unroll
    for (int l = 0; l < 3; ++l) {
#pragma unroll
      for (int q = 0; q < 4; ++q) {  // RY(params[l][q]) on qubit q
        float cc = lc[l * 4 + q], ss = ls[l * 4 + q];
        int w = 8 >> q;
#pragma unroll
        for (int i = 0; i < 16; ++i) {
          if (i & w) continue;
          int j = i | w;
          float u0 = st[i], u1 = st[j];
          st[i] = cc * u0 - ss * u1;
          st[j] = ss * u0 + cc * u1;
        }
      }
#pragma unroll
      for (int q = 0; q < 3; ++q) {  // CNOT(q, q+1): pure register permutation
        int wc = 8 >> q, wt = 4 >> q;
#pragma unroll
        for (int i = 0; i < 16; ++i) {
          if ((i & wc) && !(i & wt)) {
            int j = i | wt;
            float tsw = st[i]; st[i] = st[j]; st[j] = tsw;
          }
        }
      }
    }
    float pb[16];
#pragma unroll
    for (int i = 0; i < 16; ++i) pb[i] = st[i] * st[i];
#pragma unroll
    for (int w2 = 0; w2 < 4; ++w2) {  // <Z_w> -> feature p*4+w
      int wb = 8 >> w2;
      float m = 0.f;
#pragma unroll
      for (int i = 0; i < 16; ++i) m += (i & wb) ? -pb[i] : pb[i];
      feat[bi][p * 4 + w2] = m;
    }
  }
  __syncthreads();

  // ---- residual 1x1 stride-2 conv added in the (4,14,14) channel-major view
  for (int f = tid; f < BPB * 784; f += 256) {
    int bi = f / 784, ff = f - bi * 784;
    int ch = ff / 196, sp = ff - ch * 196;
    int i2 = sp / 14, j2 = sp - i2 * 14;
    feat[bi][ff] += x[(b0 + bi) * 784 + (2 * i2) * 28 + 2 * j2] * res_w[ch];
  }
  __syncthreads();

  // ---- classifier head: (16 x 784) @ (784 x 16 zero-padded) on wave 0 ------
  // A 16x4 f32 layout: lane<16 -> M=lane, K={k,k+1}; lane>=16 -> K={k+2,k+3}.
  // B 4x16 f32 layout: N = lane&15, same K split. D: lanes<16 M=0..7, else 8..15.
  if (tid < 32) {
    const int lane = tid;
    const int col = lane & 15;
    const int koff = (lane >= 16) ? 2 : 0;
    const bool bv = col < 10;
    const float* wrow = cls_w + (bv ? col : 0) * 784;  // clamp to stay in-bounds
    v8f acc = {};
    for (int k = 0; k < 784; k += 4) {
      v2f av, bvv;
      av.x = feat[col][k + koff];
      av.y = feat[col][k + koff + 1];
      float w0 = wrow[k + koff];
      float w1 = wrow[k + koff + 1];
      bvv.x = bv ? w0 : 0.0f;
      bvv.y = bv ? w1 : 0.0f;
      acc = __builtin_amdgcn_wmma_f32_16x16x4_f32(false, av, false, bvv,
                                                  (short)0, acc, false, false);
    }
#pragma unroll
    for (int r = 0; r < 8; ++r) {
      int M = ((lane >= 16) ? 8 : 0) + r;
      lgt[M][col] = acc[r];
    }
  }
  __syncthreads();

  // ---- bias + log_softmax, one thread per batch row ----
  if (tid < BPB) {
    float lg[10];
    float mv = -1e30f;
#pragma unroll
    for (int k = 0; k < 10; ++k) {
      lg[k] = lgt[tid][k] + cls_b[k];
      mv = fmaxf(mv, lg[k]);
    }
    float se = 0.f;
#pragma unroll
    for (int k = 0; k < 10; ++k) se += __expf(lg[k] - mv);
    float lse = mv + __logf(se);
    float* op = out + (b0 + tid) * 10;
#pragma unroll
    for (int k = 0; k < 10; ++k) op[k] = lg[k] - lse;
  }
}

extern "C" void kernel_launch(void* const* d_in, const int* in_sizes, int n_in,
                              void* d_out, int out_size, void* d_ws, size_t ws_size,
                              hipStream_t stream) {
  const float* x      = (const float*)d_in[0];  // (8192,1,28,28)
  const float* params = (const float*)d_in[1];  // (3,4,1)
  const float* res_w  = (const float*)d_in[2];  // (4,1,1,1)
  const float* cls_w  = (const float*)d_in[3];  // (10,784)
  const float* cls_b  = (const float*)d_in[4];  // (10,)
  float* out = (float*)d_out;                   // (8192,10)

  int* mnb = (int*)d_ws;        // 196 ints (float bits)
  int* mxb = mnb + POS;         // 196 ints (float bits); 1.6 KB of ws used

  qv_init<<<1, 256, 0, stream>>>(mnb, mxb);
  qv_minmax<<<64, 256, 0, stream>>>(x, mnb, mxb);
  qv_main<<<NBLK, 256, 0, stream>>>(x, params, res_w, cls_w, cls_b,
                                    (const float*)mnb, (const float*)mxb, out);
}